// detrmamba_8366596292832
// MI455X (gfx1250) — compile-verified
//
#include <hip/hip_runtime.h>
#include <math.h>

typedef __attribute__((ext_vector_type(2))) float v2f;
typedef __attribute__((ext_vector_type(8))) float v8f;

#define NB 8        // batch
#define NC 3        // c1
#define SP 4096     // 64*64 spatial
#define LS 4096     // small-mamba sequence length
#define DIS 6       // small d_inner
#define NSTATE 16   // d_state
#define DM3 4096    // m3 d_model
#define DI3 8192    // m3 d_inner
#define KXP 288     // m3 xproj rows (256 + 32)
#define MR 24       // real GEMM rows (B*L = 8*3)
#define MP 32       // padded GEMM rows (two 16-row WMMA tiles)

static __device__ __forceinline__ float silu_f(float v) {
    return v * (1.0f / (1.0f + __expf(-v)));
}
static __device__ __forceinline__ float softplus_f(float v) {
    return (v > 20.0f) ? v : log1pf(__expf(v));
}

// ---------------------------------------------------------------- generic zero fill (pad rows of A matrices)
__global__ void k_zerofill(float* __restrict__ p, int n) {
    int idx = blockIdx.x * blockDim.x + threadIdx.x;
    if (idx < n) p[idx] = 0.0f;
}

// ---------------------------------------------------------------- conv block
__global__ void k_conv1_relu(const float* __restrict__ x, const float* __restrict__ w,
                             const float* __restrict__ b, float* __restrict__ h1) {
    int idx = blockIdx.x * blockDim.x + threadIdx.x;
    if (idx >= NB * 16 * SP) return;
    int s = idx & (SP - 1);
    int oc = (idx >> 12) & 15;
    int bb = idx >> 16;
    int yy = s >> 6, xx = s & 63;
    float acc = b[oc];
    for (int ic = 0; ic < 3; ++ic)
        for (int dy = 0; dy < 3; ++dy) {
            int yi = yy + dy - 1;
            if ((unsigned)yi >= 64u) continue;
            for (int dx = 0; dx < 3; ++dx) {
                int xi = xx + dx - 1;
                if ((unsigned)xi >= 64u) continue;
                acc += w[((oc * 3 + ic) * 3 + dy) * 3 + dx] *
                       x[(bb * 3 + ic) * SP + yi * 64 + xi];
            }
        }
    h1[idx] = fmaxf(acc, 0.0f);
}

__global__ void k_conv2_res(const float* __restrict__ h1, const float* __restrict__ w,
                            const float* __restrict__ b, const float* __restrict__ x,
                            float* __restrict__ r) {
    int idx = blockIdx.x * blockDim.x + threadIdx.x;
    if (idx >= NB * NC * SP) return;
    int bb = idx / (NC * SP);
    int rem = idx - bb * (NC * SP);
    int oc = rem >> 12;
    int s = rem & (SP - 1);
    int yy = s >> 6, xx = s & 63;
    float acc = b[oc];
    for (int ic = 0; ic < 16; ++ic)
        for (int dy = 0; dy < 3; ++dy) {
            int yi = yy + dy - 1;
            if ((unsigned)yi >= 64u) continue;
            for (int dx = 0; dx < 3; ++dx) {
                int xi = xx + dx - 1;
                if ((unsigned)xi >= 64u) continue;
                acc += w[((oc * 16 + ic) * 3 + dy) * 3 + dx] *
                       h1[(bb * 16 + ic) * SP + yi * 64 + xi];
            }
        }
    r[idx] = acc + x[idx];
}

// ---------------------------------------------------------------- layernorm (over c1=3, raw reshape)
__global__ void k_layernorm(const float* __restrict__ r, const float* __restrict__ g,
                            const float* __restrict__ be, float* __restrict__ xs) {
    int idx = blockIdx.x * blockDim.x + threadIdx.x;   // (b, l)
    if (idx >= NB * LS) return;
    const float* p = r + (size_t)idx * 3;
    float a0 = p[0], a1 = p[1], a2 = p[2];
    float mu = (a0 + a1 + a2) * (1.0f / 3.0f);
    float d0 = a0 - mu, d1 = a1 - mu, d2 = a2 - mu;
    float var = (d0 * d0 + d1 * d1 + d2 * d2) * (1.0f / 3.0f);
    float inv = rsqrtf(var + 1e-5f);
    float* q = xs + (size_t)idx * 3;
    q[0] = d0 * inv * g[0] + be[0];
    q[1] = d1 * inv * g[1] + be[1];
    q[2] = d2 * inv * g[2] + be[2];
}

// ---------------------------------------------------------------- small mamba (d_model=3): stage
__global__ void k_mamba_s_stage(const float* __restrict__ xs, const float* __restrict__ in_w,
                                const float* __restrict__ cw, const float* __restrict__ cb,
                                const float* __restrict__ xp, const float* __restrict__ dtw,
                                const float* __restrict__ dtb,
                                float* __restrict__ Ust, float* __restrict__ Zst,
                                float* __restrict__ Dst, float* __restrict__ Bst,
                                float* __restrict__ Cst) {
    int idx = blockIdx.x * blockDim.x + threadIdx.x;   // (b, l)
    if (idx >= NB * LS) return;
    int bb = idx >> 12;
    int l = idx & (LS - 1);
    float xi[4][DIS];
    for (int j = 0; j < 4; ++j) {
        int lj = l - 3 + j;
        if (lj < 0) {
            for (int e = 0; e < DIS; ++e) xi[j][e] = 0.0f;
        } else {
            const float* xv = xs + (size_t)(bb * LS + lj) * 3;
            for (int e = 0; e < DIS; ++e)
                xi[j][e] = in_w[e * 3] * xv[0] + in_w[e * 3 + 1] * xv[1] + in_w[e * 3 + 2] * xv[2];
        }
    }
    const float* xv = xs + (size_t)idx * 3;
    float u[DIS], zsv[DIS];
    for (int e = 0; e < DIS; ++e) {
        float xc = cb[e];
        for (int j = 0; j < 4; ++j) xc += cw[e * 4 + j] * xi[j][e];
        u[e] = silu_f(xc);
        float zv = in_w[(DIS + e) * 3] * xv[0] + in_w[(DIS + e) * 3 + 1] * xv[1] +
                   in_w[(DIS + e) * 3 + 2] * xv[2];
        zsv[e] = silu_f(zv);
    }
    float dt = 0.0f;
    for (int e = 0; e < DIS; ++e) dt += xp[e] * u[e];                 // xproj row 0
    for (int n = 0; n < NSTATE; ++n) {
        float bm = 0.0f, cm = 0.0f;
        for (int e = 0; e < DIS; ++e) {
            bm += xp[(1 + n) * DIS + e] * u[e];
            cm += xp[(1 + NSTATE + n) * DIS + e] * u[e];
        }
        Bst[(size_t)idx * NSTATE + n] = bm;
        Cst[(size_t)idx * NSTATE + n] = cm;
    }
    for (int d = 0; d < DIS; ++d) {
        Dst[(size_t)idx * DIS + d] = softplus_f(dt * dtw[d] + dtb[d]);
        Ust[(size_t)idx * DIS + d] = u[d];
        Zst[(size_t)idx * DIS + d] = zsv[d];
    }
}

// ---------------------------------------------------------------- small mamba: serial scan, 96 thr / batch
__global__ void k_mamba_s_scan(const float* __restrict__ Ust, const float* __restrict__ Zst,
                               const float* __restrict__ Dst, const float* __restrict__ Bst,
                               const float* __restrict__ Cst, const float* __restrict__ A_log,
                               const float* __restrict__ Dp, const float* __restrict__ out_w,
                               float* __restrict__ out, int mode) {
    __shared__ float ylds[DIS];
    int bb = blockIdx.x;
    int t = threadIdx.x;                 // 0..95
    int d = t >> 4, n = t & 15;
    float A_dn = -__expf(A_log[d * NSTATE + n]);
    float h = 0.0f;
    float ow0 = (t < 3) ? out_w[t * DIS + 0] : 0.0f;
    float ow1 = (t < 3) ? out_w[t * DIS + 1] : 0.0f;
    float ow2 = (t < 3) ? out_w[t * DIS + 2] : 0.0f;
    float ow3 = (t < 3) ? out_w[t * DIS + 3] : 0.0f;
    float ow4 = (t < 3) ? out_w[t * DIS + 4] : 0.0f;
    float ow5 = (t < 3) ? out_w[t * DIS + 5] : 0.0f;
    for (int l = 0; l < LS; ++l) {
        size_t base = (size_t)(bb * LS + l);
        float delta = Dst[base * DIS + d];
        float uu = Ust[base * DIS + d];
        float Bm = Bst[base * NSTATE + n];
        float Cm = Cst[base * NSTATE + n];
        float dA = __expf(delta * A_dn);
        h = dA * h + delta * Bm * uu;
        float c = h * Cm;
        for (int off = 8; off > 0; off >>= 1) c += __shfl_xor(c, off, 16);
        if (n == 0) {
            float y = (c + uu * Dp[d]) * Zst[base * DIS + d];
            ylds[d] = y;
        }
        __syncthreads();
        if (t < 3) {
            float o = ow0 * ylds[0] + ow1 * ylds[1] + ow2 * ylds[2] +
                      ow3 * ylds[3] + ow4 * ylds[4] + ow5 * ylds[5];
            if (mode == 0)
                out[(size_t)(bb * 3 + t) * LS + l] = o;          // y1: (B,3,L) == padded A
            else
                out[base * 3 + t] = o;                            // att_pre: (B,L,3)
        }
        __syncthreads();
    }
}

// ---------------------------------------------------------------- skinny fp32 WMMA GEMM
// C[MR x N] = A[MP x K] * Wt[N x K]^T ; A is zero-padded to MP=32 rows, so ALL loads
// are unconditional (no EXEC masking in the hot loop). One wave per 16-wide N tile,
// both 16-row M tiles carried per wave => weight stream read exactly once.
__global__ void k_gemm_skinny_wmma(const float* __restrict__ A, int lda,
                                   const float* __restrict__ Wt, int K,
                                   float* __restrict__ C, int ldc,
                                   const float* __restrict__ bias, int act) {
    int lane = threadIdx.x;              // 32 lanes
    int n0 = blockIdx.x << 4;
    int mr = lane & 15;
    int koff = (lane >> 4) << 1;         // 0 or 2: K-pair selector per f32 WMMA layout
    const float* wrow  = Wt + (size_t)(n0 + mr) * K + koff;
    const float* arow0 = A + (size_t)mr * lda + koff;
    const float* arow1 = A + (size_t)(mr + 16) * lda + koff;
    v8f acc0 = {};
    v8f acc1 = {};
    for (int kb = 0; kb < K; kb += 64) {
        __builtin_prefetch(wrow + kb + 512, 0, 3);   // global_prefetch_b8, ~2KB ahead in weight stream
#pragma unroll
        for (int ki = 0; ki < 64; ki += 4) {
            v2f bf = *(const v2f*)(wrow + kb + ki);
            v2f a0 = *(const v2f*)(arow0 + kb + ki);
            v2f a1 = *(const v2f*)(arow1 + kb + ki);
            acc0 = __builtin_amdgcn_wmma_f32_16x16x4_f32(false, a0, false, bf,
                                                         (short)0, acc0, false, false);
            acc1 = __builtin_amdgcn_wmma_f32_16x16x4_f32(false, a1, false, bf,
                                                         (short)0, acc1, false, false);
        }
    }
    int col = n0 + mr;
    int rbase = (lane >> 4) << 3;        // D layout: row = v + 8*(lane>>4)
    float bv = (bias != nullptr) ? bias[col] : 0.0f;
#pragma unroll
    for (int v = 0; v < 8; ++v) {
        int row = rbase + v;
        if (row < MR) {
            float x0 = acc0[v];
            if (act) x0 = softplus_f(x0 + bv);
            C[(size_t)row * ldc + col] = x0;
        }
        int row1 = row + 16;
        if (row1 < MR) {
            float x1 = acc1[v];
            if (act) x1 = softplus_f(x1 + bv);
            C[(size_t)row1 * ldc + col] = x1;
        }
    }
}

// ---------------------------------------------------------------- m3 causal depthwise conv (L=3) + silu, z-gate
__global__ void k_m3_convsilu(const float* __restrict__ xz3, const float* __restrict__ cw,
                              const float* __restrict__ cb, float* __restrict__ u3,
                              float* __restrict__ zs3) {
    int idx = blockIdx.x * blockDim.x + threadIdx.x;   // (b, e)
    if (idx >= NB * DI3) return;
    int bb = idx >> 13;
    int e = idx & (DI3 - 1);
    size_t r0 = (size_t)(bb * 3 + 0) * (2 * DI3);
    size_t r1 = (size_t)(bb * 3 + 1) * (2 * DI3);
    size_t r2 = (size_t)(bb * 3 + 2) * (2 * DI3);
    float xi0 = xz3[r0 + e], xi1 = xz3[r1 + e], xi2 = xz3[r2 + e];
    float w1 = cw[e * 4 + 1], w2 = cw[e * 4 + 2], w3 = cw[e * 4 + 3];
    float bias = cb[e];
    u3[(size_t)(bb * 3 + 0) * DI3 + e] = silu_f(w3 * xi0 + bias);
    u3[(size_t)(bb * 3 + 1) * DI3 + e] = silu_f(w2 * xi0 + w3 * xi1 + bias);
    u3[(size_t)(bb * 3 + 2) * DI3 + e] = silu_f(w1 * xi0 + w2 * xi1 + w3 * xi2 + bias);
    zs3[(size_t)(bb * 3 + 0) * DI3 + e] = silu_f(xz3[r0 + DI3 + e]);
    zs3[(size_t)(bb * 3 + 1) * DI3 + e] = silu_f(xz3[r1 + DI3 + e]);
    zs3[(size_t)(bb * 3 + 2) * DI3 + e] = silu_f(xz3[r2 + DI3 + e]);
}

// ---------------------------------------------------------------- m3 scan (L=3, one thread per (b,d))
__global__ void k_m3_scan(const float* __restrict__ delta3, const float* __restrict__ u3,
                          const float* __restrict__ zs3, const float* __restrict__ xdbl3,
                          const float* __restrict__ A_log3, const float* __restrict__ D3,
                          float* __restrict__ yg3) {
    int idx = blockIdx.x * blockDim.x + threadIdx.x;
    if (idx >= NB * DI3) return;
    int bb = idx >> 13;
    int d = idx & (DI3 - 1);
    float An[NSTATE];
    for (int n = 0; n < NSTATE; ++n) An[n] = -__expf(A_log3[d * NSTATE + n]);
    float h[NSTATE];
    for (int n = 0; n < NSTATE; ++n) h[n] = 0.0f;
    float Dd = D3[d];
    for (int l = 0; l < 3; ++l) {
        size_t row = (size_t)(bb * 3 + l);
        float delta = delta3[row * DI3 + d];
        float uu = u3[row * DI3 + d];
        const float* xd = xdbl3 + row * KXP;
        float y = 0.0f;
        for (int n = 0; n < NSTATE; ++n) {
            float dA = __expf(delta * An[n]);
            h[n] = dA * h[n] + delta * xd[256 + n] * uu;
            y += h[n] * xd[272 + n];
        }
        y += uu * Dd;
        yg3[row * DI3 + d] = y * zs3[row * DI3 + d];
    }
}

// ---------------------------------------------------------------- softmax over batch (legacy dim=0) * z3
__global__ void k_softmax_combine(const float* __restrict__ attp, const float* __restrict__ out3,
                                  float* __restrict__ comb) {
    int idx = blockIdx.x * blockDim.x + threadIdx.x;   // l*3 + o, 12288 total
    if (idx >= LS * 3) return;
    int l = idx / 3;
    int o = idx - l * 3;
    float v[NB];
    float mx = -3.4e38f;
    for (int b = 0; b < NB; ++b) {
        v[b] = attp[(size_t)(b * LS + l) * 3 + o];
        mx = fmaxf(mx, v[b]);
    }
    float s = 0.0f;
    for (int b = 0; b < NB; ++b) {
        v[b] = __expf(v[b] - mx);
        s += v[b];
    }
    float inv = 1.0f / s;
    for (int b = 0; b < NB; ++b) {
        float z3 = out3[(size_t)(b * 3 + o) * DM3 + l];   // z3[b][l][o] = out3[b][o][l]
        comb[(size_t)b * (NC * SP) + idx] = v[b] * inv * z3;
    }
}

// ---------------------------------------------------------------- smooth conv (3->3, 3x3)
__global__ void k_smooth_conv(const float* __restrict__ comb, const float* __restrict__ w,
                              const float* __restrict__ b, float* __restrict__ out) {
    int idx = blockIdx.x * blockDim.x + threadIdx.x;
    if (idx >= NB * NC * SP) return;
    int bb = idx / (NC * SP);
    int rem = idx - bb * (NC * SP);
    int oc = rem >> 12;
    int s = rem & (SP - 1);
    int yy = s >> 6, xx = s & 63;
    float acc = b[oc];
    for (int ic = 0; ic < 3; ++ic)
        for (int dy = 0; dy < 3; ++dy) {
            int yi = yy + dy - 1;
            if ((unsigned)yi >= 64u) continue;
            for (int dx = 0; dx < 3; ++dx) {
                int xi = xx + dx - 1;
                if ((unsigned)xi >= 64u) continue;
                acc += w[((oc * 3 + ic) * 3 + dy) * 3 + dx] *
                       comb[(size_t)bb * (NC * SP) + ic * SP + yi * 64 + xi];
            }
        }
    out[idx] = acc;
}

// ---------------------------------------------------------------- launcher
extern "C" void kernel_launch(void* const* d_in, const int* in_sizes, int n_in,
                              void* d_out, int out_size, void* d_ws, size_t ws_size,
                              hipStream_t stream) {
    (void)in_sizes; (void)n_in; (void)out_size; (void)ws_size;
    const float* x    = (const float*)d_in[0];
    const float* cw1  = (const float*)d_in[1];
    const float* cb1  = (const float*)d_in[2];
    const float* cw2  = (const float*)d_in[3];
    const float* cb2  = (const float*)d_in[4];
    const float* lng  = (const float*)d_in[5];
    const float* lnb  = (const float*)d_in[6];
    // m1: 7..15, m2: 16..24, m3: 25..33  (in_w, conv_w, conv_b, xproj_w, dt_w, dt_b, A_log, D, out_w)
    const float* m1_inw = (const float*)d_in[7];
    const float* m1_cw  = (const float*)d_in[8];
    const float* m1_cb  = (const float*)d_in[9];
    const float* m1_xp  = (const float*)d_in[10];
    const float* m1_dtw = (const float*)d_in[11];
    const float* m1_dtb = (const float*)d_in[12];
    const float* m1_al  = (const float*)d_in[13];
    const float* m1_D   = (const float*)d_in[14];
    const float* m1_ow  = (const float*)d_in[15];
    const float* m2_inw = (const float*)d_in[16];
    const float* m2_cw  = (const float*)d_in[17];
    const float* m2_cb  = (const float*)d_in[18];
    const float* m2_xp  = (const float*)d_in[19];
    const float* m2_dtw = (const float*)d_in[20];
    const float* m2_dtb = (const float*)d_in[21];
    const float* m2_al  = (const float*)d_in[22];
    const float* m2_D   = (const float*)d_in[23];
    const float* m2_ow  = (const float*)d_in[24];
    const float* m3_inw = (const float*)d_in[25];
    const float* m3_cw  = (const float*)d_in[26];
    const float* m3_cb  = (const float*)d_in[27];
    const float* m3_xp  = (const float*)d_in[28];
    const float* m3_dtw = (const float*)d_in[29];
    const float* m3_dtb = (const float*)d_in[30];
    const float* m3_al  = (const float*)d_in[31];
    const float* m3_D   = (const float*)d_in[32];
    const float* m3_ow  = (const float*)d_in[33];
    const float* smw   = (const float*)d_in[34];
    const float* smb   = (const float*)d_in[35];
    float* out = (float*)d_out;

    float* ws = (float*)d_ws;
    size_t off = 0;
    float* h1    = ws + off; off += (size_t)NB * 16 * SP;        // 524288
    float* rbuf  = ws + off; off += (size_t)NB * NC * SP;        // 98304
    float* xs    = ws + off; off += (size_t)NB * LS * 3;         // 98304
    float* Ust   = ws + off; off += (size_t)NB * LS * DIS;       // 196608
    float* Zst   = ws + off; off += (size_t)NB * LS * DIS;
    float* Dst   = ws + off; off += (size_t)NB * LS * DIS;
    float* Bst   = ws + off; off += (size_t)NB * LS * NSTATE;    // 524288
    float* Cst   = ws + off; off += (size_t)NB * LS * NSTATE;
    float* y1    = ws + off; off += (size_t)MP * DM3;            // padded A (32 x 4096)
    float* attp  = ws + off; off += (size_t)NB * LS * 3;         // 98304
    float* xz3   = ws + off; off += (size_t)MR * 2 * DI3;        // 393216 (C only, 24 rows)
    float* u3    = ws + off; off += (size_t)MP * DI3;            // padded A (32 x 8192)
    float* zs3   = ws + off; off += (size_t)MR * DI3;
    float* xdbl3 = ws + off; off += (size_t)MP * KXP;            // padded A (32 x 288)
    float* delta3= ws + off; off += (size_t)MR * DI3;
    float* yg3   = ws + off; off += (size_t)MP * DI3;            // padded A (32 x 8192)
    float* out3  = ws + off; off += (size_t)MR * DM3;            // 98304
    float* comb  = ws + off; off += (size_t)NB * NC * SP;        // 98304

    const int BS = 256;
    // zero the 8 pad rows of every WMMA A-matrix (keeps the GEMM hot loop branch-free)
    k_zerofill<<<(8 * DM3 + BS - 1) / BS, BS, 0, stream>>>(y1 + (size_t)MR * DM3, 8 * DM3);
    k_zerofill<<<(8 * DI3 + BS - 1) / BS, BS, 0, stream>>>(u3 + (size_t)MR * DI3, 8 * DI3);
    k_zerofill<<<(8 * KXP + BS - 1) / BS, BS, 0, stream>>>(xdbl3 + (size_t)MR * KXP, 8 * KXP);
    k_zerofill<<<(8 * DI3 + BS - 1) / BS, BS, 0, stream>>>(yg3 + (size_t)MR * DI3, 8 * DI3);
    // conv block + layernorm
    k_conv1_relu<<<(NB * 16 * SP + BS - 1) / BS, BS, 0, stream>>>(x, cw1, cb1, h1);
    k_conv2_res<<<(NB * NC * SP + BS - 1) / BS, BS, 0, stream>>>(h1, cw2, cb2, x, rbuf);
    k_layernorm<<<(NB * LS + BS - 1) / BS, BS, 0, stream>>>(rbuf, lng, lnb, xs);
    // m1 (-> y1 rows 0..23, (B,3,4096) == GEMM A rows)
    k_mamba_s_stage<<<(NB * LS + BS - 1) / BS, BS, 0, stream>>>(
        xs, m1_inw, m1_cw, m1_cb, m1_xp, m1_dtw, m1_dtb, Ust, Zst, Dst, Bst, Cst);
    k_mamba_s_scan<<<NB, 96, 0, stream>>>(Ust, Zst, Dst, Bst, Cst, m1_al, m1_D, m1_ow, y1, 0);
    // m2 (-> att_pre, reuses staging buffers; stream-ordered after m1 scan)
    k_mamba_s_stage<<<(NB * LS + BS - 1) / BS, BS, 0, stream>>>(
        xs, m2_inw, m2_cw, m2_cb, m2_xp, m2_dtw, m2_dtb, Ust, Zst, Dst, Bst, Cst);
    k_mamba_s_scan<<<NB, 96, 0, stream>>>(Ust, Zst, Dst, Bst, Cst, m2_al, m2_D, m2_ow, attp, 1);
    // m3: four skinny WMMA GEMMs (M=24 padded to 32) around conv/scan/gating
    k_gemm_skinny_wmma<<<(2 * DI3) / 16, 32, 0, stream>>>(
        y1, DM3, m3_inw, DM3, xz3, 2 * DI3, nullptr, 0);                 // in-proj (N=16384,K=4096)
    k_m3_convsilu<<<(NB * DI3 + BS - 1) / BS, BS, 0, stream>>>(xz3, m3_cw, m3_cb, u3, zs3);
    k_gemm_skinny_wmma<<<KXP / 16, 32, 0, stream>>>(
        u3, DI3, m3_xp, DI3, xdbl3, KXP, nullptr, 0);                    // x_dbl (N=288,K=8192)
    k_gemm_skinny_wmma<<<DI3 / 16, 32, 0, stream>>>(
        xdbl3, KXP, m3_dtw, 256, delta3, DI3, m3_dtb, 1);               // delta (N=8192,K=256,softplus)
    k_m3_scan<<<(NB * DI3 + BS - 1) / BS, BS, 0, stream>>>(
        delta3, u3, zs3, xdbl3, m3_al, m3_D, yg3);
    k_gemm_skinny_wmma<<<DM3 / 16, 32, 0, stream>>>(
        yg3, DI3, m3_ow, DI3, out3, DM3, nullptr, 0);                    // out-proj (N=4096,K=8192)
    // softmax(dim=0) * z3, reshape, smooth conv
    k_softmax_combine<<<(LS * 3 + BS - 1) / BS, BS, 0, stream>>>(attp, out3, comb);
    k_smooth_conv<<<(NB * NC * SP + BS - 1) / BS, BS, 0, stream>>>(comb, smw, smb, out);
}